// FFTConv1d_thresh_31860067402453
// MI455X (gfx1250) — compile-verified
//
#include <hip/hip_runtime.h>
#include <hip/hip_bf16.h>

typedef __attribute__((ext_vector_type(2))) float v2f;
typedef __attribute__((ext_vector_type(8))) float v8f;
typedef __attribute__((ext_vector_type(4))) unsigned int u32x4;
typedef __attribute__((ext_vector_type(8))) int i32x8;
typedef __attribute__((ext_vector_type(4))) int i32x4;

#define LD144 144
#define MS144 (144 * 144)           // one padded 144x144 matrix (floats)
#define NIMG 128                     // 4 batches * 32 channels
#define NVALID 130                   // true DFT length per axis (130*130 = 16900)
#define KPAD 132                     // K extent (33 WMMA steps of 4)

// ---------------------------------------------------------------------------
// WMMA helper: D = A(16x4) * B(4x16) + C, fp32 via V_WMMA_F32_16X16X4_F32
// ---------------------------------------------------------------------------
static __device__ inline v8f wmma4(v2f a, v2f b, v8f c) {
  return __builtin_amdgcn_wmma_f32_16x16x4_f32(
      /*neg_a=*/false, a, /*neg_b=*/false, b,
      /*c_mod=*/(short)0, c, /*reuse_a=*/false, /*reuse_b=*/false);
}

// ---------------------------------------------------------------------------
// TDM: DMA a 2D tile (tile_w x tile_h elems, fp32) from a strided global
// tensor into LDS (rows packed contiguously). Descriptor per CDNA5 ISA ch.8:
//   group0: count=1 | lds_addr | global_addr(57b) | type=2
//   group1: data_size=4B, tensor dims, tile dims, dim0 stride
//   groups 2/3 (+trailing group): zero (2D tensor)
// 6-arg builtin form (clang-23 / therock-10.0 headers).
// Issued once per calling wave; completion tracked by TENSORcnt.
// ---------------------------------------------------------------------------
static __device__ inline void tdm_load_2d(const float* gsrc, float* lds_dst,
                                          int tile_w, int tile_h,
                                          int tensor_w, int tensor_h,
                                          int stride_elems) {
  const unsigned lds_off = (unsigned)(uintptr_t)lds_dst;            // LDS byte offset
  const unsigned long long ga = (unsigned long long)(uintptr_t)gsrc;
  u32x4 g0;
  g0.x = 1u;                                                        // count=1 (valid)
  g0.y = lds_off;                                                   // lds_addr
  g0.z = (unsigned)(ga & 0xFFFFFFFFull);                            // global_addr[31:0]
  g0.w = (unsigned)((ga >> 32) & 0x01FFFFFFull) | (2u << 30);       // ga[56:32] | type=2
  i32x8 g1;
  g1[0] = (2 << 16);                                                // data_size=4B
  g1[1] = (tensor_w & 0xFFFF) << 16;                                // tensor_dim0[15:0]
  g1[2] = ((tensor_w >> 16) & 0xFFFF) | ((tensor_h & 0xFFFF) << 16);// dim0 hi | dim1 lo
  g1[3] = ((tensor_h >> 16) & 0xFFFF) | ((tile_w & 0xFFFF) << 16);  // dim1 hi | tile_dim0
  g1[4] = (tile_h & 0xFFFF);                                        // tile_dim1 (tile_dim2=0)
  g1[5] = stride_elems;                                             // tensor_dim0_stride[31:0]
  g1[6] = 0;
  g1[7] = 0;
  const i32x4 z4 = {0, 0, 0, 0};
  const i32x8 z8 = {0, 0, 0, 0, 0, 0, 0, 0};
  __builtin_amdgcn_tensor_load_to_lds(g0, g1, z4, z4, z8, 0);
}

// ---------------------------------------------------------------------------
// Build DFT / twiddle tables (zero-padded to 144x144).
// ---------------------------------------------------------------------------
__global__ void build_tables_kernel(float* __restrict__ Ffr, float* __restrict__ Ffi,
                                    float* __restrict__ Fir, float* __restrict__ Fii,
                                    float* __restrict__ TWfr, float* __restrict__ TWfi,
                                    float* __restrict__ TWir, float* __restrict__ TWii) {
  int idx = blockIdx.x * blockDim.x + threadIdx.x;
  if (idx >= MS144) return;
  int r = idx / LD144, c = idx % LD144;
  float fr = 0.f, fi = 0.f, gr = 0.f, gi = 0.f;
  float tfr = 0.f, tfi = 0.f, tir = 0.f, tii = 0.f;
  if (r < NVALID && c < NVALID) {
    const float TWO_PI = 6.28318530717958647692f;
    int m1 = (r * c) % NVALID;                      // exact phase reduction
    float a = TWO_PI * (float)m1 / (float)NVALID;
    float ca = cosf(a), sa = sinf(a);
    fr = ca; fi = -sa;                              // forward DFT matrix
    gr = ca; gi = sa;                               // inverse (unscaled)
    long long m2 = ((long long)r * (long long)c) % 16900ll;
    float b = TWO_PI * (float)m2 / 16900.0f;
    float cb = cosf(b), sb = sinf(b);
    tfr = cb; tfi = -sb;                            // forward twiddle
    tir = cb; tii = sb;                             // inverse twiddle
  }
  Ffr[idx] = fr;  Ffi[idx] = fi;
  Fir[idx] = gr;  Fii[idx] = gi;
  TWfr[idx] = tfr; TWfi[idx] = tfi;
  TWir[idx] = tir; TWii[idx] = tii;
}

// ---------------------------------------------------------------------------
// Zero-pad input: x (img,128,128) -> XP (144x144), data at rows/cols 1..128.
// ---------------------------------------------------------------------------
__global__ void pad_input_kernel(const float* __restrict__ x, float* __restrict__ XP) {
  int img = blockIdx.y;
  int idx = blockIdx.x * blockDim.x + threadIdx.x;
  if (idx >= MS144) return;
  int r = idx / LD144, c = idx % LD144;
  float v = 0.f;
  if (r >= 1 && r <= 128 && c >= 1 && c <= 128)
    v = x[(size_t)img * 16384 + (size_t)(r - 1) * 128 + (c - 1)];
  XP[(size_t)img * MS144 + idx] = v;
}

// ---------------------------------------------------------------------------
// Batched 144x144x132 complex GEMM on the f32 WMMA pipe, compile-time
// specialized:
//   BC: B is complex          DC: D is complex (else real part only)
//   TW: fuse D *= TW[row][col] (complex twiddle) into epilogue
//   TH: fuse spectral pruning (|Re|<0.01 -> 0) into epilogue
// Block 32x9 (9 waves): wave ty owns output rows [16*ty,16*ty+16).
// The full A panel (144x132 re+im, 152 KB) is DMA'd into LDS by the Tensor
// Data Mover (CDNA5 WGP has 320 KB LDS) while threads stage the first B panel;
// B panels (132x16) are staged per N tile, transposed so K pairs are
// contiguous ds_load_b64. Inner loop is pure LDS + WMMA.
// ---------------------------------------------------------------------------
template <bool BC, bool DC, bool TW, bool TH>
__global__ void cgemm144_kernel(const float* __restrict__ Are, const float* __restrict__ Aim,
                                const float* __restrict__ Bre, const float* __restrict__ Bim,
                                float* __restrict__ Dre, float* __restrict__ Dim,
                                const float* __restrict__ TWre, const float* __restrict__ TWim,
                                int aPerImg, int bPerImg, float scale) {
  __shared__ float sAre[LD144 * KPAD];            // 76 KB  (sA[r*KPAD + k])
  __shared__ float sAim[LD144 * KPAD];            // 76 KB
  __shared__ float sBre[16 * KPAD];               // 8.4 KB (sB[c*KPAD + k])
  __shared__ float sBim[BC ? 16 * KPAD : 1];

  const int img = blockIdx.x;
  const size_t aOff = (size_t)(aPerImg ? img : 0) * MS144;
  const size_t bOff = (size_t)(bPerImg ? img : 0) * MS144;
  const float* __restrict__ ar = Are + aOff;
  const float* __restrict__ ai = Aim + aOff;
  const float* __restrict__ br = Bre + bOff;
  const float* __restrict__ bi = BC ? (Bim + bOff) : nullptr;
  float* __restrict__ dr = Dre + (size_t)img * MS144;
  float* __restrict__ di = DC ? (Dim + (size_t)img * MS144) : nullptr;

  const int lane = threadIdx.x;        // 0..31
  const int g = lane >> 4;             // half-wave select (K pair / row-half)
  const int m = lane & 15;             // row within A tile / col within B,D tile
  const int mt16 = (int)threadIdx.y * 16;
  const int tid = (int)threadIdx.y * 32 + lane;

  // ---- TDM: DMA the full A panel (re+im) into LDS; wave 0 issues both ----
  if (threadIdx.y == 0) {
    tdm_load_2d(ar, sAre, KPAD, LD144, LD144, LD144, LD144);
    tdm_load_2d(ai, sAim, KPAD, LD144, LD144, LD144, LD144);
  }

  // Per-lane LDS bases.
  const float* __restrict__ saRe = sAre + (mt16 + m) * KPAD + 2 * g;
  const float* __restrict__ saIm = sAim + (mt16 + m) * KPAD + 2 * g;
  const float* __restrict__ sbRe = sBre + m * KPAD + 2 * g;
  const float* __restrict__ sbIm = sBim + m * KPAD + 2 * g;

  for (int nt16 = 0; nt16 < LD144; nt16 += 16) {
    // ---- cooperative stage of B panel (rows 0..131, cols nt16..nt16+15) ----
    __syncthreads();                    // previous tile's readers done
    for (int i = tid; i < 16 * KPAD; i += 32 * 9) {
      const int k = i >> 4;             // 0..131
      const int c = i & 15;
      sBre[c * KPAD + k] = br[k * LD144 + nt16 + c];
      if (BC) sBim[c * KPAD + k] = bi[k * LD144 + nt16 + c];
    }
    // A-panel DMA must be complete before the first MACs; overlapped with
    // the nt==0 B staging above.
    if (nt16 == 0 && threadIdx.y == 0) __builtin_amdgcn_s_wait_tensorcnt(0);
    __syncthreads();

    v8f cre = {0.f, 0.f, 0.f, 0.f, 0.f, 0.f, 0.f, 0.f};
    v8f cim = {0.f, 0.f, 0.f, 0.f, 0.f, 0.f, 0.f, 0.f};

#pragma unroll 3
    for (int k0 = 0; k0 < KPAD; k0 += 4) {
      const v2f a_re = *(const v2f*)(saRe + k0);
      const v2f a_im = *(const v2f*)(saIm + k0);
      const v2f b_re = *(const v2f*)(sbRe + k0);
      cre = wmma4(a_re, b_re, cre);                 // += Ar*Br
      if (BC) {
        const v2f b_im = *(const v2f*)(sbIm + k0);
        cre = wmma4(a_im, -b_im, cre);              // += -Ai*Bi
        if (DC) cim = wmma4(a_re, b_im, cim);       // += Ar*Bi
      }
      if (DC) cim = wmma4(a_im, b_re, cim);         // += Ai*Br
    }

    // ---- epilogue: optional twiddle / threshold, then store ----
    union { v8f v; float f[8]; } ur, ui;
    ur.v = cre; ui.v = cim;
    const int dcol = nt16 + m;
#pragma unroll
    for (int j = 0; j < 8; ++j) {
      const int row = mt16 + 8 * g + j;
      float re = ur.f[j] * scale;
      float im = DC ? ui.f[j] * scale : 0.f;
      if (TW) {
        const float wr = TWre[row * LD144 + dcol];
        const float wi = TWim[row * LD144 + dcol];
        const float r2 = re * wr - im * wi;
        im = re * wi + im * wr;
        re = r2;
      }
      if (TH) {
        if (fabsf(re) < 0.01f) { re = 0.f; im = 0.f; }
      }
      dr[row * LD144 + dcol] = re;
      if (DC) di[row * LD144 + dcol] = im;
    }
  }
}

// ---------------------------------------------------------------------------
// Final 3x3 correlation over 32 channels on the filtered padded images + bias.
// grid = (64, O=32, B=4), block = 256 (1 output pixel / thread).
// ---------------------------------------------------------------------------
__global__ void conv3x3_kernel(const float* __restrict__ XT, const float* __restrict__ w,
                               const float* __restrict__ bias, float* __restrict__ out) {
  __shared__ float sW[32 * 9];
  const int o = blockIdx.y;
  const int b = blockIdx.z;
  for (int i = threadIdx.x; i < 32 * 9; i += blockDim.x) sW[i] = w[o * 32 * 9 + i];
  __syncthreads();

  const int pix = blockIdx.x * blockDim.x + threadIdx.x;   // 0..16383
  const int r = pix >> 7, c = pix & 127;
  float acc = bias[o];
#pragma unroll 4
  for (int ci = 0; ci < 32; ++ci) {
    const float* p = XT + (size_t)(b * 32 + ci) * MS144 + (size_t)r * LD144 + c;
    const float* k = sW + ci * 9;
    acc += k[0] * p[0]         + k[1] * p[1]           + k[2] * p[2]
         + k[3] * p[LD144]     + k[4] * p[LD144 + 1]   + k[5] * p[LD144 + 2]
         + k[6] * p[2 * LD144] + k[7] * p[2 * LD144+1] + k[8] * p[2 * LD144+2];
  }
  out[(((size_t)b * 32 + o) * 128 + r) * 128 + c] = acc;
}

// ---------------------------------------------------------------------------
// Orchestration. Inputs: x (4,32,128,128) f32, weight (32,32,3,3) f32, bias (32).
// ---------------------------------------------------------------------------
extern "C" void kernel_launch(void* const* d_in, const int* in_sizes, int n_in,
                              void* d_out, int out_size, void* d_ws, size_t ws_size,
                              hipStream_t stream) {
  const float* x    = (const float*)d_in[0];
  const float* wgt  = (const float*)d_in[1];
  const float* bias = (const float*)d_in[2];
  float* out = (float*)d_out;
  float* ws  = (float*)d_ws;

  // Workspace layout (floats): 8 tables + XP + 2 complex ping-pong buffers + XT.
  float* Ffr  = ws + (size_t)0 * MS144;
  float* Ffi  = ws + (size_t)1 * MS144;
  float* Fir  = ws + (size_t)2 * MS144;
  float* Fii  = ws + (size_t)3 * MS144;
  float* TWfr = ws + (size_t)4 * MS144;
  float* TWfi = ws + (size_t)5 * MS144;
  float* TWir = ws + (size_t)6 * MS144;
  float* TWii = ws + (size_t)7 * MS144;
  float* XP   = ws + (size_t)8 * MS144;                      // 128 real images
  float* B1r  = XP  + (size_t)NIMG * MS144;
  float* B1i  = B1r + (size_t)NIMG * MS144;
  float* B2r  = B1i + (size_t)NIMG * MS144;
  float* B2i  = B2r + (size_t)NIMG * MS144;
  float* XT   = B2i + (size_t)NIMG * MS144;                  // filtered real images

  const dim3 gemm_grid(NIMG), gemm_block(32, 9);

  build_tables_kernel<<<(MS144 + 255) / 256, 256, 0, stream>>>(Ffr, Ffi, Fir, Fii,
                                                               TWfr, TWfi, TWir, TWii);
  pad_input_kernel<<<dim3((MS144 + 255) / 256, NIMG), 256, 0, stream>>>(x, XP);

  // Forward stage 1 (+fused fwd twiddle):  T2 = (Ffwd x XP) .* TWf
  cgemm144_kernel<false, true, true, false><<<gemm_grid, gemm_block, 0, stream>>>(
      Ffr, Ffi, XP, nullptr, B1r, B1i, TWfr, TWfi, 0, 1, 1.0f);
  // Forward stage 2 (+fused pruning):      S = prune(T2 x Ffwd)
  cgemm144_kernel<true, true, false, true><<<gemm_grid, gemm_block, 0, stream>>>(
      B1r, B1i, Ffr, Ffi, B2r, B2i, nullptr, nullptr, 1, 0, 1.0f);
  // Inverse stage 1 (+fused inv twiddle):  U2 = (S x Finv) .* TWi
  cgemm144_kernel<true, true, true, false><<<gemm_grid, gemm_block, 0, stream>>>(
      B2r, B2i, Fir, Fii, B1r, B1i, TWir, TWii, 1, 0, 1.0f);
  // Inverse stage 2 (real part, 1/N):      XT = Re(Finv x U2) / 16900
  cgemm144_kernel<true, false, false, false><<<gemm_grid, gemm_block, 0, stream>>>(
      Fir, Fii, B1r, B1i, XT, nullptr, nullptr, nullptr, 0, 1, 1.0f / 16900.0f);

  // 3x3 correlation over channels + bias.
  conv3x3_kernel<<<dim3(64, 32, 4), 256, 0, stream>>>(XT, wgt, bias, out);
}